// CombinedLossWithEMD_38448547234598
// MI455X (gfx1250) — compile-verified
//
#include <hip/hip_runtime.h>
#include <math.h>

typedef __attribute__((ext_vector_type(2))) float v2f;
typedef __attribute__((ext_vector_type(8))) float v8f;

#define DIST_EPS 1e-12f
#define REP_THRESH 0.01f

// workspace accumulator layout (floats):
// acc[0] = sum of row-mins pred->gt
// acc[1] = sum of row-mins gt->pred
// acc[2] = sum of row-mins partial->pred   (coverage)
// acc[3] = repulsion sum
// acc[4] = smoothness (variance) sum
__global__ void cl_init_acc(float* acc) {
    if (threadIdx.x < 8) acc[threadIdx.x] = 0.0f;
}

// One 16x16 tile of SQUARED distances, computed entirely by the matrix unit:
//   D = (-2*X) x Y^T + C,  C[m][n] = |x_m|^2 + |y_n|^2   =>  D[m][n] = d^2(m,n)
// Am2 holds the A operand pre-scaled by -2; asq_bc[] holds |x|^2 broadcast to
// the D-tile row layout. EPS clamp is deferred (max(.,eps) commutes with
// min / top-k), as is sqrt (monotonic).
__device__ __forceinline__ v8f dist2_tile(const float* __restrict__ Yb, int n0,
                                          int ml, bool hiHalf, v2f Am2,
                                          const float* asq_bc) {
    const float bx = Yb[(n0 + ml) * 3 + 0];
    const float by = Yb[(n0 + ml) * 3 + 1];
    const float bz = Yb[(n0 + ml) * 3 + 2];
    v2f Bm;
    Bm.x = hiHalf ? bz : bx;
    Bm.y = hiHalf ? 0.0f : by;
    const float bsq = bx * bx + by * by + bz * bz;  // |y_n|^2, n = lane&15

    v8f C;
#pragma unroll
    for (int r = 0; r < 8; ++r) C[r] = asq_bc[r] + bsq;
    return __builtin_amdgcn_wmma_f32_16x16x4_f32(false, Am2, false, Bm,
                                                 (short)0, C, false, false);
}

// Loads this wave's 16 X rows, builds the -2-scaled A operand and the |x|^2
// broadcast table matching the D-tile layout.
__device__ __forceinline__ void load_A(const float* __restrict__ Xb, int m0,
                                       int ml, bool hiHalf, v2f* Am2,
                                       float* asq_bc) {
    const float ax = Xb[(m0 + ml) * 3 + 0];
    const float ay = Xb[(m0 + ml) * 3 + 1];
    const float az = Xb[(m0 + ml) * 3 + 2];
    // A operand 16x4 f32: lanes 0-15: K0,K1 ; lanes 16-31: K2,K3(pad 0)
    Am2->x = -2.0f * (hiHalf ? az : ax);
    Am2->y = hiHalf ? 0.0f : (-2.0f * ay);
    const float asq_self = ax * ax + ay * ay + az * az;  // m = lane&15
#pragma unroll
    for (int r = 0; r < 8; ++r)
        asq_bc[r] = __shfl(asq_self, hiHalf ? (r + 8) : r, 32);
}

// For each row x of X[b] (N points), min over all M points of Y[b] of the
// euclidean distance; atomically accumulates the sum of mins into acc[slot].
// One wave32 per 16-row block; minimisation runs on raw squared distances,
// clamp+sqrt applied to the 8 survivors.
__global__ __launch_bounds__(256) void cl_rowmin_wmma(
    const float* __restrict__ X, const float* __restrict__ Y,
    int N, int M, float* __restrict__ acc, int slot) {
    const int lane = threadIdx.x & 31;
    const int waveInBlock = threadIdx.x >> 5;
    const int rowBlock = blockIdx.x * (blockDim.x >> 5) + waveInBlock;
    const int b = blockIdx.y;
    const int m0 = rowBlock * 16;
    if (m0 >= N) return;  // wave-uniform: EXEC stays all-ones for WMMA

    const float* __restrict__ Xb = X + (size_t)b * N * 3;
    const float* __restrict__ Yb = Y + (size_t)b * M * 3;
    const int  ml = lane & 15;
    const bool hiHalf = lane >= 16;

    v2f Am2;
    float asq_bc[8];
    load_A(Xb, m0, ml, hiHalf, &Am2, asq_bc);

    float rmin[8];
#pragma unroll
    for (int r = 0; r < 8; ++r) rmin[r] = 3.0e38f;

    // main loop: 2 column tiles (2 WMMAs) per iteration
    int n0 = 0;
    for (; n0 + 32 <= M; n0 += 32) {
        __builtin_prefetch(&Yb[(n0 + 32 + ml) * 3], 0, 3);
        const v8f d2a = dist2_tile(Yb, n0,      ml, hiHalf, Am2, asq_bc);
        const v8f d2b = dist2_tile(Yb, n0 + 16, ml, hiHalf, Am2, asq_bc);
#pragma unroll
        for (int r = 0; r < 8; ++r)  // lowers to v_min3_num_f32
            rmin[r] = fminf(rmin[r], fminf(d2a[r], d2b[r]));
    }
    for (; n0 < M; n0 += 16) {  // safety tail (unused for these sizes)
        const v8f d2a = dist2_tile(Yb, n0, ml, hiHalf, Am2, asq_bc);
#pragma unroll
        for (int r = 0; r < 8; ++r) rmin[r] = fminf(rmin[r], d2a[r]);
    }

    // reduce squared mins across the 16 lanes of each half; clamp+sqrt once
    float s = 0.0f;
#pragma unroll
    for (int r = 0; r < 8; ++r) {
        float v = rmin[r];
#pragma unroll
        for (int off = 1; off < 16; off <<= 1)
            v = fminf(v, __shfl_xor(v, off, 32));
        s += __builtin_amdgcn_sqrtf(fmaxf(v, DIST_EPS));  // lane0: m=r, lane16: m=r+8
    }
    if (lane == 0 || lane == 16) atomicAdd(&acc[slot], s);
}

// pred x pred: per-row 16 smallest squared distances (self included) via WMMA
// tiles staged through LDS; lanes 0-15 each own one row's sorted top-16 in
// registers; clamp+sqrt applied to the 16 survivors at the end.
__global__ __launch_bounds__(32) void cl_knn_wmma(
    const float* __restrict__ P, int N, float* __restrict__ acc) {
    __shared__ float tile[32][17];  // two 16x16 tiles, +1 pad vs bank conflicts
    const int lane = threadIdx.x;
    const int b = blockIdx.y;
    const int m0 = blockIdx.x * 16;
    const float* __restrict__ Pb = P + (size_t)b * N * 3;
    const int  ml = lane & 15;
    const bool hiHalf = lane >= 16;

    v2f Am2;
    float asq_bc[8];
    load_A(Pb, m0, ml, hiHalf, &Am2, asq_bc);

    float t[16];
#pragma unroll
    for (int i = 0; i < 16; ++i) t[i] = 3.0e38f;

    for (int n0 = 0; n0 < N; n0 += 32) {
        __builtin_prefetch(&Pb[(n0 + 32 + ml) * 3], 0, 3);
        const v8f d2a = dist2_tile(Pb, n0,      ml, hiHalf, Am2, asq_bc);
        const v8f d2b = dist2_tile(Pb, n0 + 16, ml, hiHalf, Am2, asq_bc);
#pragma unroll
        for (int r = 0; r < 8; ++r) {
            const int row = hiHalf ? (r + 8) : r;
            tile[row][ml]      = d2a[r];
            tile[row + 16][ml] = d2b[r];
        }
        __syncthreads();
        if (lane < 16) {
#pragma unroll
            for (int n = 0; n < 32; ++n) {
                float v = tile[(n & 1) ? (lane + 16) : lane][n >> 1];
                // early reject: after warm-up almost no candidate survives
                if (v < t[15]) {
                    // branch-free sorted insert: t ascending, drops the max
#pragma unroll
                    for (int i = 0; i < 16; ++i) {
                        const float lo = fminf(t[i], v);
                        const float hv = fmaxf(t[i], v);
                        t[i] = lo;
                        v = hv;
                    }
                }
            }
        }
        __syncthreads();
    }

    if (lane < 16) {
#pragma unroll
        for (int i = 0; i < 16; ++i)
            t[i] = __builtin_amdgcn_sqrtf(fmaxf(t[i], DIST_EPS));
        // repulsion: k = 1..4 (t[0] is the self distance)
        float rep = 0.0f;
#pragma unroll
        for (int k = 1; k <= 4; ++k) rep += fmaxf(REP_THRESH - t[k], 0.0f);
        // smoothness: unbiased variance of the 16 NN distances
        float mean = 0.0f;
#pragma unroll
        for (int i = 0; i < 16; ++i) mean += t[i];
        mean *= (1.0f / 16.0f);
        float var = 0.0f;
#pragma unroll
        for (int i = 0; i < 16; ++i) {
            const float dm = t[i] - mean;
            var = fmaf(dm, dm, var);
        }
        var *= (1.0f / 15.0f);
        atomicAdd(&acc[3], rep);
        atomicAdd(&acc[4], var);
    }
}

__global__ void cl_finalize(const float* __restrict__ acc,
                            float* __restrict__ out, int B, int N, int Kp) {
    if (threadIdx.x == 0 && blockIdx.x == 0) {
        const float cd     = acc[0] / (float)(B * N) + acc[1] / (float)(B * N);
        const float cov    = acc[2] / (float)(B * Kp);
        const float rep    = acc[3] / (float)(B * N * 4);
        const float smooth = acc[4] / (float)(B * N);
        const float total  = 1.0f * cd + 0.01f * rep + 0.005f * smooth + 0.1f * cov;
        out[0] = total;
        out[1] = cd;
        out[2] = rep;
        out[3] = smooth;
        out[4] = cov;
    }
}

extern "C" void kernel_launch(void* const* d_in, const int* in_sizes, int n_in,
                              void* d_out, int out_size, void* d_ws, size_t ws_size,
                              hipStream_t stream) {
    const float* pred    = (const float*)d_in[0];
    const float* gt      = (const float*)d_in[1];
    const float* partial = (const float*)d_in[2];

    const int B  = 4;
    const int N  = in_sizes[0] / (B * 3);  // 4096
    const int M  = in_sizes[1] / (B * 3);  // 4096
    const int Kp = in_sizes[2] / (B * 3);  // 2048

    float* acc = (float*)d_ws;
    float* out = (float*)d_out;

    cl_init_acc<<<1, 32, 0, stream>>>(acc);

    const int wavesPerBlock = 8;  // 256 threads = 8 wave32, one row-block each
    cl_rowmin_wmma<<<dim3((N / 16 + wavesPerBlock - 1) / wavesPerBlock, B),
                     dim3(256), 0, stream>>>(pred, gt, N, M, acc, 0);
    cl_rowmin_wmma<<<dim3((M / 16 + wavesPerBlock - 1) / wavesPerBlock, B),
                     dim3(256), 0, stream>>>(gt, pred, M, N, acc, 1);
    cl_rowmin_wmma<<<dim3((Kp / 16 + wavesPerBlock - 1) / wavesPerBlock, B),
                     dim3(256), 0, stream>>>(partial, pred, Kp, N, acc, 2);

    cl_knn_wmma<<<dim3(N / 16, B), dim3(32), 0, stream>>>(pred, N, acc);

    cl_finalize<<<1, 1, 0, stream>>>(acc, out, B, N, Kp);
}